// RNNDecoder_52106543235628
// MI455X (gfx1250) — compile-verified
//
#include <hip/hip_runtime.h>
#include <hip/hip_bf16.h>

typedef __attribute__((ext_vector_type(2))) float v2f;
typedef __attribute__((ext_vector_type(8))) float v8f;

static constexpr int PAD_ID = 50256;
static constexpr int B = 32, T = 64, S = 400;
static constexpr int D = 512, DK = 64;   // DW=DM=DE=D

// ---- d_out layout (floats, concatenated in reference return order) ----
static constexpr int OFF_PRED  = 0;                      // [B,T,D/2]
static constexpr int OFF_ATTN  = OFF_PRED + B*T*(D/2);   // [B,S]   (attns[-1])
static constexpr int OFF_CTXF  = OFF_ATTN + B*S;         // [B,D]
static constexpr int OFF_CPRED = OFF_CTXF + B*D;         // [B,T,S]
static constexpr int OFF_GATE  = OFF_CPRED + B*T*S;      // [B,T,1]
static constexpr int OFF_COVL  = OFF_GATE + B*T;         // [B,T]

// ---- workspace layout (floats) ----
static constexpr int WS_PRE = 0;                 // precompute [B*S, DK]
static constexpr int WS_H0  = WS_PRE + B*S*DK;   // h buffer 0 [B,D]
static constexpr int WS_H1  = WS_H0 + B*D;       // h buffer 1 [B,D]
static constexpr int WS_CTX = WS_H1 + B*D;       // ctx [B,D]
static constexpr int WS_COV = WS_CTX + B*D;      // cov [B,S]  (contiguous after ctx)

__device__ __forceinline__ v8f wmma4(v2f a, v2f b, v8f c) {
  // D = A(16x4,f32) * B(4x16,f32) + C(16x16,f32)
  return __builtin_amdgcn_wmma_f32_16x16x4_f32(false, a, false, b, (short)0, c,
                                               false, false);
}
__device__ __forceinline__ float sigm(float x) { return 1.0f / (1.0f + expf(-x)); }

// -------------------------------------------------------------------------
__global__ void zero_kernel(float* p, int n) {
  int i = blockIdx.x * blockDim.x + threadIdx.x;
  if (i < n) p[i] = 0.0f;
}

// h0 = tanh(hidden @ W_init + b_init) : [32 x 512], K=512. 32 blocks x 2 waves.
__global__ void h0_kernel(const float* __restrict__ hidden,
                          const float* __restrict__ Wi,
                          const float* __restrict__ bi,
                          float* __restrict__ h0) {
  const int lane = threadIdx.x & 31, wave = threadIdx.x >> 5;
  const int lc = lane & 15, half = lane >> 4, koff = half * 2;
  const int m0 = wave * 16;
  const int colB = blockIdx.x * 16 + lc;
  const float* ar = hidden + (size_t)(m0 + lc) * D;
  v8f acc = {};
  for (int k0 = 0; k0 < D; k0 += 4) {
    v2f a; a.x = ar[k0 + koff]; a.y = ar[k0 + koff + 1];
    const float* w0 = Wi + (size_t)(k0 + koff) * D + colB;
    v2f b; b.x = w0[0]; b.y = w0[D];
    acc = wmma4(a, b, acc);
  }
  const int rbase = m0 + half * 8;
  const float bias = bi[colB];
#pragma unroll
  for (int i = 0; i < 8; ++i)
    h0[(size_t)(rbase + i) * D + colB] = tanhf(acc[i] + bias);
}

// precompute = enc_output @ Wc : [B*S x 512] @ [512 x 64]. 800 blocks x 4 waves.
__global__ void pre_kernel(const float* __restrict__ enc,
                           const float* __restrict__ Wc,
                           float* __restrict__ pre) {
  const int lane = threadIdx.x & 31, wave = threadIdx.x >> 5;
  const int lc = lane & 15, half = lane >> 4, koff = half * 2;
  const int m0 = blockIdx.x * 16;        // row tile in flattened [B*S]
  const int colB = wave * 16 + lc;       // 4 waves cover DK=64
  const float* ar = enc + (size_t)(m0 + lc) * D;
  v8f acc = {};
  for (int k0 = 0; k0 < D; k0 += 4) {
    v2f a; a.x = ar[k0 + koff]; a.y = ar[k0 + koff + 1];
    const float* w0 = Wc + (size_t)(k0 + koff) * DK + colB;
    v2f b; b.x = w0[0]; b.y = w0[DK];
    acc = wmma4(a, b, acc);
  }
  const int rbase = m0 + half * 8;
#pragma unroll
  for (int i = 0; i < 8; ++i)
    pre[(size_t)(rbase + i) * DK + colB] = acc[i];
}

// GRU step: h_out = GRUCell([emb(t), ctx], h_in). 32 blocks x 64 thr (2 waves).
// Each wave: one 16x16 tile of each of the 6 gate pre-activations via WMMA.
__global__ void gru_kernel(const int* __restrict__ tgt,
                           const float* __restrict__ wemb,
                           const float* __restrict__ ctx,
                           const float* __restrict__ h_in,
                           const float* __restrict__ Wx,
                           const float* __restrict__ Wh,
                           const float* __restrict__ bx,
                           const float* __restrict__ bh,
                           float* __restrict__ h_out, int t) {
  const int lane = threadIdx.x & 31, wave = threadIdx.x >> 5;
  const int lc = lane & 15, half = lane >> 4, koff = half * 2;
  const int m0 = wave * 16;
  const int colB = blockIdx.x * 16 + lc;      // gate column j in [0,512)
  const int rowA = m0 + lc;                   // batch row for A-operand loads

  const int tok = tgt[rowA * T + t];
  const float* ep = wemb + (size_t)tok * D;   // embedding row (on-the-fly gather)
  const float* cp = ctx + (size_t)rowA * D;
  const float* hp = h_in + (size_t)rowA * D;

  v8f xr = {}, xz = {}, xn = {};   // x-path gate accumulators
  v8f gr = {}, gz = {}, gn = {};   // h-path gate accumulators

  // gx part 1: K = 0..511 from embedding, Wx rows [0,512)
  for (int k0 = 0; k0 < D; k0 += 4) {
    v2f a; a.x = ep[k0 + koff]; a.y = ep[k0 + koff + 1];
    const float* w0 = Wx + (size_t)(k0 + koff) * 1536 + colB;
    v2f b0; b0.x = w0[0];    b0.y = w0[1536];
    v2f b1; b1.x = w0[512];  b1.y = w0[512 + 1536];
    v2f b2; b2.x = w0[1024]; b2.y = w0[1024 + 1536];
    xr = wmma4(a, b0, xr);
    xz = wmma4(a, b1, xz);
    xn = wmma4(a, b2, xn);
  }
  // gx part 2: K = 512..1023 from ctx (input feed), Wx rows [512,1024)
  for (int k0 = 0; k0 < D; k0 += 4) {
    v2f a; a.x = cp[k0 + koff]; a.y = cp[k0 + koff + 1];
    const float* w0 = Wx + (size_t)(D + k0 + koff) * 1536 + colB;
    v2f b0; b0.x = w0[0];    b0.y = w0[1536];
    v2f b1; b1.x = w0[512];  b1.y = w0[512 + 1536];
    v2f b2; b2.x = w0[1024]; b2.y = w0[1024 + 1536];
    xr = wmma4(a, b0, xr);
    xz = wmma4(a, b1, xz);
    xn = wmma4(a, b2, xn);
  }
  // gh: K = 0..511 from h_in, Wh rows [0,512)
  for (int k0 = 0; k0 < D; k0 += 4) {
    v2f a; a.x = hp[k0 + koff]; a.y = hp[k0 + koff + 1];
    const float* w0 = Wh + (size_t)(k0 + koff) * 1536 + colB;
    v2f b0; b0.x = w0[0];    b0.y = w0[1536];
    v2f b1; b1.x = w0[512];  b1.y = w0[512 + 1536];
    v2f b2; b2.x = w0[1024]; b2.y = w0[1024 + 1536];
    gr = wmma4(a, b0, gr);
    gz = wmma4(a, b1, gz);
    gn = wmma4(a, b2, gn);
  }

  const float bxr = bx[colB], bxz = bx[512 + colB], bxn = bx[1024 + colB];
  const float bhr = bh[colB], bhz = bh[512 + colB], bhn = bh[1024 + colB];
  const int rbase = m0 + half * 8;
#pragma unroll
  for (int i = 0; i < 8; ++i) {
    const int r = rbase + i;
    float rg = sigm(xr[i] + bxr + gr[i] + bhr);
    float zg = sigm(xz[i] + bxz + gz[i] + bhz);
    float ng = tanhf(xn[i] + bxn + rg * (gn[i] + bhn));
    float hv = h_in[(size_t)r * D + colB];
    h_out[(size_t)r * D + colB] = (1.0f - zg) * ng + zg * hv;
  }
}

// Attention + softmax + coverage + context. One block per batch row.
__global__ void attn_kernel(const int* __restrict__ src,
                            const float* __restrict__ enc,
                            const float* __restrict__ pre,
                            const float* __restrict__ Wq,
                            const float* __restrict__ wcov,
                            const float* __restrict__ vatt,
                            const float* __restrict__ h,
                            float* __restrict__ ctx,
                            float* __restrict__ cov,
                            float* __restrict__ out, int t) {
  __shared__ float hq_s[DK], wc_s[DK], va_s[DK];
  __shared__ float sc_s[S], at_s[S];
  __shared__ float red[256];
  const int b = blockIdx.x;
  const int tid = threadIdx.x;

  if (tid < DK) {                         // hq = h[b] @ Wq  (coalesced over k)
    float acc = 0.0f;
    for (int d = 0; d < D; ++d) acc += h[b * D + d] * Wq[d * DK + tid];
    hq_s[tid] = acc;
    wc_s[tid] = wcov[tid];
    va_s[tid] = vatt[tid];
  }
  __syncthreads();

  float lmax = -3.0e38f;
  for (int s = tid; s < S; s += 256) {
    float cv = cov[b * S + s];
    const float* pr = pre + (size_t)(b * S + s) * DK;
    float e = 0.0f;
    for (int k = 0; k < DK; ++k)
      e += tanhf(pr[k] + hq_s[k] + cv * wc_s[k]) * va_s[k];
    if (src[b * S + s] == PAD_ID) e = -1e9f;
    sc_s[s] = e;
    lmax = fmaxf(lmax, e);
  }
  red[tid] = lmax; __syncthreads();
  for (int o = 128; o > 0; o >>= 1) {
    if (tid < o) red[tid] = fmaxf(red[tid], red[tid + o]);
    __syncthreads();
  }
  const float m = red[0];
  __syncthreads();

  float lsum = 0.0f;
  for (int s = tid; s < S; s += 256) {
    float p = expf(sc_s[s] - m);
    at_s[s] = p;
    lsum += p;
  }
  red[tid] = lsum; __syncthreads();
  for (int o = 128; o > 0; o >>= 1) {
    if (tid < o) red[tid] += red[tid + o];
    __syncthreads();
  }
  const float inv = 1.0f / red[0];
  __syncthreads();

  const float tden = 1.0f / fmaxf((float)t, 1.0f);
  float lcl = 0.0f;
  for (int s = tid; s < S; s += 256) {
    float a = at_s[s] * inv;
    at_s[s] = a;
    out[OFF_CPRED + ((b * T) + t) * S + s] = a;
    if (t == T - 1) out[OFF_ATTN + b * S + s] = a;
    float cv = cov[b * S + s];
    lcl += fminf(a, cv * tden);
    cov[b * S + s] = cv + a;
  }
  red[tid] = lcl; __syncthreads();
  for (int o = 128; o > 0; o >>= 1) {
    if (tid < o) red[tid] += red[tid + o];
    __syncthreads();
  }
  if (tid == 0) out[OFF_COVL + b * T + t] = red[0];
  __syncthreads();

  // new_ctx[b,d] = sum_s attn[s] * enc[b,s,d]  (coalesced across lanes in d)
  for (int d = tid; d < D; d += 256) {
    float acc = 0.0f;
    const float* e0 = enc + (size_t)(b * S) * D + d;
    for (int s = 0; s < S; ++s) acc += at_s[s] * e0[(size_t)s * D];
    ctx[b * D + d] = acc;
    if (t == T - 1) out[OFF_CTXF + b * D + d] = acc;
  }
}

// Readout (maxout-2) + copy gate. 33 blocks x 64 thr: 32 WMMA tile blocks + 1 gate block.
__global__ void readout_kernel(const int* __restrict__ tgt,
                               const float* __restrict__ wemb,
                               const float* __restrict__ h,
                               const float* __restrict__ ctx,
                               const float* __restrict__ Wr,
                               const float* __restrict__ br,
                               const float* __restrict__ Wcp,
                               const float* __restrict__ bcp,
                               float* __restrict__ out, int t) {
  if (blockIdx.x == 32) {                 // copy gate (tiny, scalar)
    const int b = threadIdx.x;
    if (b < B) {
      float acc = bcp[0];
      for (int k = 0; k < D; ++k) acc += h[b * D + k] * Wcp[k];
      for (int k = 0; k < D; ++k) acc += ctx[b * D + k] * Wcp[D + k];
      out[OFF_GATE + b * T + t] = sigm(acc);
    }
    return;
  }
  const int lane = threadIdx.x & 31, wave = threadIdx.x >> 5;
  const int lc = lane & 15, half = lane >> 4, koff = half * 2;
  const int m0 = wave * 16;
  const int colB = blockIdx.x * 16 + lc;
  const int rowA = m0 + lc;

  const int tok = tgt[rowA * T + t];
  const float* ep = wemb + (size_t)tok * D;
  const float* hp = h + (size_t)rowA * D;
  const float* cp = ctx + (size_t)rowA * D;

  v8f acc = {};
  for (int k0 = 0; k0 < D; k0 += 4) {           // K block 1: embedding
    v2f a; a.x = ep[k0 + koff]; a.y = ep[k0 + koff + 1];
    const float* w0 = Wr + (size_t)(k0 + koff) * D + colB;
    v2f b; b.x = w0[0]; b.y = w0[D];
    acc = wmma4(a, b, acc);
  }
  for (int k0 = 0; k0 < D; k0 += 4) {           // K block 2: h
    v2f a; a.x = hp[k0 + koff]; a.y = hp[k0 + koff + 1];
    const float* w0 = Wr + (size_t)(D + k0 + koff) * D + colB;
    v2f b; b.x = w0[0]; b.y = w0[D];
    acc = wmma4(a, b, acc);
  }
  for (int k0 = 0; k0 < D; k0 += 4) {           // K block 3: ctx
    v2f a; a.x = cp[k0 + koff]; a.y = cp[k0 + koff + 1];
    const float* w0 = Wr + (size_t)(2 * D + k0 + koff) * D + colB;
    v2f b; b.x = w0[0]; b.y = w0[D];
    acc = wmma4(a, b, acc);
  }

  const float bias = br[colB];
  const int rbase = m0 + half * 8;
#pragma unroll
  for (int i = 0; i < 8; ++i) {
    float v = acc[i] + bias;
    float o = __shfl_xor(v, 1, 32);             // partner column (maxout pair)
    float mx = fmaxf(v, o);
    if ((colB & 1) == 0)
      out[OFF_PRED + ((rbase + i) * T + t) * (D / 2) + (colB >> 1)] = mx;
  }
}

// -------------------------------------------------------------------------
extern "C" void kernel_launch(void* const* d_in, const int* in_sizes, int n_in,
                              void* d_out, int out_size, void* d_ws, size_t ws_size,
                              hipStream_t stream) {
  (void)in_sizes; (void)n_in; (void)out_size; (void)ws_size;
  const int*   tgt  = (const int*)d_in[0];
  const int*   src  = (const int*)d_in[1];
  const float* enc  = (const float*)d_in[2];
  const float* hid  = (const float*)d_in[3];
  const float* wemb = (const float*)d_in[4];
  const float* Wi   = (const float*)d_in[5];
  const float* bi   = (const float*)d_in[6];
  const float* Wx   = (const float*)d_in[7];
  const float* Wh   = (const float*)d_in[8];
  const float* bx   = (const float*)d_in[9];
  const float* bh   = (const float*)d_in[10];
  const float* Wc   = (const float*)d_in[11];
  const float* Wq   = (const float*)d_in[12];
  const float* wcov = (const float*)d_in[13];
  const float* vatt = (const float*)d_in[14];
  const float* Wcp  = (const float*)d_in[15];
  const float* bcp  = (const float*)d_in[16];
  const float* Wr   = (const float*)d_in[17];
  const float* br   = (const float*)d_in[18];
  float* out = (float*)d_out;
  float* ws  = (float*)d_ws;

  float* pre   = ws + WS_PRE;
  float* hb[2] = { ws + WS_H0, ws + WS_H1 };
  float* ctx   = ws + WS_CTX;
  float* cov   = ws + WS_COV;

  // ctx and cov are contiguous: zero both in one launch (fresh state every call)
  zero_kernel<<<(B * D + B * S + 255) / 256, 256, 0, stream>>>(ctx, B * D + B * S);
  h0_kernel<<<D / 16, 64, 0, stream>>>(hid, Wi, bi, hb[0]);
  pre_kernel<<<(B * S) / 16, 128, 0, stream>>>(enc, Wc, pre);

  for (int t = 0; t < T; ++t) {
    float* hin  = hb[t & 1];
    float* hout = hb[(t + 1) & 1];
    gru_kernel<<<D / 16, 64, 0, stream>>>(tgt, wemb, ctx, hin, Wx, Wh, bx, bh,
                                          hout, t);
    attn_kernel<<<B, 256, 0, stream>>>(src, enc, pre, Wq, wcov, vatt, hout, ctx,
                                       cov, out, t);
    readout_kernel<<<D / 16 + 1, 64, 0, stream>>>(tgt, wemb, hout, ctx, Wr, br,
                                                  Wcp, bcp, out, t);
  }
}